// SpectralConv3d_63960652972650
// MI455X (gfx1250) — compile-verified
//
#include <hip/hip_runtime.h>
#include <hip/hip_bf16.h>

typedef float v2f __attribute__((ext_vector_type(2)));
typedef float v8f __attribute__((ext_vector_type(8)));

// ---------------------------------------------------------------------------
// Generic batched GEMM against a small twiddle matrix, using f32 WMMA 16x16x4.
// A is addressed as  A[(r/Q)*SA + k*KA + (r%Q)]  (r = logical row)
// D is addressed as  out[(r/Qo)*SO + col*Co + (r%Qo)]
// Twiddle matrices stored column-major per matrix: tw[mat*Npad*K + col*K + k]
// MODE 0 (real -> complex):  out0 = A*cos, out1 = A*(-sin)      (mats: c, -s)
// MODE 1 (cplx -> cplx):     Re = Ar*c + Ai*s ; Im = -Ar*s+Ai*c (mats: c, s, -s)
// MODE 2 (cplx -> Re-Im):    out0 = Ar*(c+s) + Ai*(s-c)         (mats: c+s, s-c)
// ---------------------------------------------------------------------------
struct StageParams {
  const float* Are;
  const float* Aim;
  float*       out0;
  float*       out1;
  const float* tw;
  int K, Npad, Nvalid;
  int rowTiles, colTiles;
  int Q, SA, KA;
  int Qo, SO, Co;
  float scale;
};

template <int MODE>
__global__ __launch_bounds__(256) void dht_gemm(StageParams p) {
  const int waveInBlk = threadIdx.x >> 5;
  const int tile = blockIdx.x * 8 + waveInBlk;
  const int tiles = p.rowTiles * p.colTiles;
  if (tile >= tiles) return;  // wave-uniform exit: EXEC all-1s for WMMA

  const int rowTile = tile / p.colTiles;
  const int colTile = tile % p.colTiles;
  const int lane = threadIdx.x & 31;
  const int half = lane >> 4;
  const int m    = lane & 15;

  const int row  = rowTile * 16 + m;          // row this lane loads for A
  const int col0 = colTile * 16 + m;          // col this lane loads/stores

  const long aBase = (long)(row / p.Q) * p.SA + (row % p.Q);
  const float* tw0 = p.tw + (long)col0 * p.K; // column-major twiddle base
  const int matStride = p.Npad * p.K;

  v8f acc0 = {};
  v8f acc1 = {};

  for (int k0 = 0; k0 < p.K; k0 += 4) {
    const int ka = k0 + 2 * half;
    v2f a0, a1, b0, b1, b2;
    const long ao = aBase + (long)ka * p.KA;
    a0.x = p.Are[ao];
    a0.y = p.Are[ao + p.KA];
    b0 = *(const v2f*)(tw0 + ka);
    if (MODE == 0) {
      b1 = *(const v2f*)(tw0 + matStride + ka);
      acc0 = __builtin_amdgcn_wmma_f32_16x16x4_f32(false, a0, false, b0,
                                                   (short)0, acc0, false, false);
      acc1 = __builtin_amdgcn_wmma_f32_16x16x4_f32(false, a0, false, b1,
                                                   (short)0, acc1, false, false);
    } else if (MODE == 1) {
      a1.x = p.Aim[ao];
      a1.y = p.Aim[ao + p.KA];
      b1 = *(const v2f*)(tw0 + matStride + ka);       // sin
      b2 = *(const v2f*)(tw0 + 2 * matStride + ka);   // -sin
      acc0 = __builtin_amdgcn_wmma_f32_16x16x4_f32(false, a0, false, b0,
                                                   (short)0, acc0, false, false);
      acc0 = __builtin_amdgcn_wmma_f32_16x16x4_f32(false, a1, false, b1,
                                                   (short)0, acc0, false, false);
      acc1 = __builtin_amdgcn_wmma_f32_16x16x4_f32(false, a0, false, b2,
                                                   (short)0, acc1, false, false);
      acc1 = __builtin_amdgcn_wmma_f32_16x16x4_f32(false, a1, false, b0,
                                                   (short)0, acc1, false, false);
    } else {
      a1.x = p.Aim[ao];
      a1.y = p.Aim[ao + p.KA];
      b1 = *(const v2f*)(tw0 + matStride + ka);       // s-c
      acc0 = __builtin_amdgcn_wmma_f32_16x16x4_f32(false, a0, false, b0,
                                                   (short)0, acc0, false, false);
      acc0 = __builtin_amdgcn_wmma_f32_16x16x4_f32(false, a1, false, b1,
                                                   (short)0, acc0, false, false);
    }
  }

  // D layout: lane<16 -> rows 0..7, lane>=16 -> rows 8..15; col = col0
  const bool colOK = col0 < p.Nvalid;
  int r   = rowTile * 16 + half * 8;
  int q   = r / p.Qo;
  int rem = r % p.Qo;
  for (int i = 0; i < 8; ++i) {
    if (colOK) {
      const long o = (long)q * p.SO + (long)col0 * p.Co + rem;
      p.out0[o] = acc0[i] * p.scale;
      if (MODE != 2) p.out1[o] = acc1[i] * p.scale;
    }
    rem++;
    if (rem == p.Qo) { rem = 0; q++; }
  }
}

// ---------------------------------------------------------------------------
// Twiddle generator. theta = 2*pi*((mapK(k)*mapC(col)) mod L)/L, exact mod-L
// reduction for accuracy. Column-major per matrix.
// ---------------------------------------------------------------------------
__global__ void twiddle_init(float* tw, int K, int Npad, int Nvalid, int L,
                             int mapK, int mapC, int mode, int nMats) {
  const int i = blockIdx.x * 256 + threadIdx.x;
  const int total = nMats * Npad * K;
  if (i >= total) return;
  const int mat = i / (Npad * K);
  const int rem = i % (Npad * K);
  const int col = rem / K;
  const int k   = rem % K;
  float v = 0.0f;
  if (col < Nvalid) {
    const int fk = mapK ? (k < 16 ? k : k + 32) : k;
    const int fc = mapC ? (col < 16 ? col : col + 32) : col;
    const int prod = (fk * fc) % L;
    const float theta = 6.28318530717958647692f * (float)prod / (float)L;
    const float c = cosf(theta);
    const float s = sinf(theta);
    if (mode == 0)      v = (mat == 0) ? c : -s;
    else if (mode == 1) v = (mat == 0) ? c : ((mat == 1) ? s : -s);
    else                v = (mat == 0) ? (c + s) : (s - c);
  }
  tw[i] = v;
}

// ---------------------------------------------------------------------------
// Channel mixing: Y[b,co,dd,hh,ww] = sum_ci X[b,ci,dd,hh,ww]*w_sel[ci,co,...]
// Weight-bandwidth bound (each weight used B=2 times) -> VALU dot products.
// One block per (dd,hh) in the 32x32 corner grid; 256 threads = (coSlot, z).
// ---------------------------------------------------------------------------
__global__ __launch_bounds__(256) void mix_kernel(const float* __restrict__ X,
                                                  const float* __restrict__ w1,
                                                  const float* __restrict__ w2,
                                                  const float* __restrict__ w3,
                                                  const float* __restrict__ w4,
                                                  float* __restrict__ Y) {
  const int dd = blockIdx.x >> 5;
  const int hh = blockIdx.x & 31;
  __shared__ float xs[2048];  // [b][ci][z]
  const int t = threadIdx.x;
  for (int i = t; i < 2048; i += 256) {
    const int b  = i >> 10;
    const int ci = (i >> 4) & 63;
    const int z  = i & 15;
    xs[i] = X[((((long)(b * 64 + ci) * 32 + dd) * 32 + hh) << 4) + z];
  }
  __syncthreads();

  const float* w = (dd < 16) ? ((hh < 16) ? w1 : w3) : ((hh < 16) ? w2 : w4);
  const int xd = dd & 15, yh = hh & 15;
  const int z = t & 15;
  const int coSlot = t >> 4;
  const int soff = (xd * 16 + yh) * 16 + z;

  for (int cq = 0; cq < 4; ++cq) {
    const int co = coSlot * 4 + cq;
    float acc0 = 0.0f, acc1 = 0.0f;
    for (int ci = 0; ci < 64; ++ci) {
      const float wv = w[(long)(ci * 64 + co) * 4096 + soff];
      acc0 += xs[ci * 16 + z] * wv;
      acc1 += xs[1024 + ci * 16 + z] * wv;
    }
    Y[((((long)(co) * 32 + dd) * 32 + hh) << 4) + z] = acc0;
    Y[((((long)(64 + co) * 32 + dd) * 32 + hh) << 4) + z] = acc1;
  }
}

// ---------------------------------------------------------------------------
static void launch_stage(int mode, const StageParams& p, hipStream_t s) {
  const int tiles = p.rowTiles * p.colTiles;
  const int blocks = (tiles + 7) / 8;
  if (mode == 0)      dht_gemm<0><<<blocks, 256, 0, s>>>(p);
  else if (mode == 1) dht_gemm<1><<<blocks, 256, 0, s>>>(p);
  else                dht_gemm<2><<<blocks, 256, 0, s>>>(p);
}

extern "C" void kernel_launch(void* const* d_in, const int* in_sizes, int n_in,
                              void* d_out, int out_size, void* d_ws, size_t ws_size,
                              hipStream_t stream) {
  const float* x  = (const float*)d_in[0];
  const float* w1 = (const float*)d_in[1];
  const float* w2 = (const float*)d_in[2];
  const float* w3 = (const float*)d_in[3];
  const float* w4 = (const float*)d_in[4];
  float* out = (float*)d_out;
  float* ws  = (float*)d_ws;

  // Workspace layout (floats)
  float* tw  = ws;                       // 6 slots * 8192
  float* A0  = ws + 49152;               // arena A: 16777216 (F1 then G1)
  float* BB  = A0 + 16777216;            // arena B: 17301504 (F2 then G2)
  float* Xb  = BB + 17301504;            // 2097152
  float* Yb  = Xb + 2097152;             // 2097152
  (void)ws_size; (void)in_sizes; (void)n_in; (void)out_size;

  float* F1re = A0, *F1im = A0 + 8388608;
  float* F2re = BB, *F2im = BB + 4194304;
  float* G1re = A0, *G1im = A0 + 8388608;
  float* G2re = BB, *G2im = BB + 8650752;

  // --- twiddle tables (tiny) ---
  // slot, K, Npad, Nvalid, L, mapK, mapC, mode, nMats
  struct TW { int K, Np, Nv, L, mK, mC, mode, nM; };
  const TW tws[6] = {
      {64, 16, 16, 64, 0, 0, 0, 2},   // FW
      {64, 32, 32, 64, 0, 1, 1, 3},   // FH
      {64, 32, 32, 64, 0, 1, 2, 2},   // FD
      {16, 48, 33, 33, 0, 0, 0, 2},   // I1
      {32, 64, 64, 64, 1, 0, 1, 3},   // I2
      {32, 64, 64, 64, 1, 0, 2, 2},   // I3
  };
  for (int s = 0; s < 6; ++s) {
    const int n = tws[s].nM * tws[s].Np * tws[s].K;
    twiddle_init<<<(n + 255) / 256, 256, 0, stream>>>(
        tw + s * 8192, tws[s].K, tws[s].Np, tws[s].Nv, tws[s].L,
        tws[s].mK, tws[s].mC, tws[s].mode, tws[s].nM);
  }

  StageParams p;

  // FW: x (b,ci,d,h,w64) -> F1 (b,ci,d,h,ww16) complex. rows=(b,ci,d,h)=524288
  p = {x, nullptr, F1re, F1im, tw + 0 * 8192,
       64, 16, 16, 32768, 1, 1, 64, 1, 1, 16, 1, 1.0f};
  launch_stage(0, p, stream);

  // FH: F1 -> F2 (b,ci,d,hh32,ww16) complex. rows=(b,ci,d,ww)=131072
  p = {F1re, F1im, F2re, F2im, tw + 1 * 8192,
       64, 32, 32, 8192, 2, 16, 1024, 16, 16, 512, 16, 1.0f};
  launch_stage(1, p, stream);

  // FD: F2 -> X (b,ci,dd32,hh32,ww16) real (Re-Im). rows=(b,ci,hh,ww)=65536
  p = {F2re, F2im, Xb, nullptr, tw + 2 * 8192,
       64, 32, 32, 4096, 2, 512, 32768, 512, 512, 16384, 512, 1.0f};
  launch_stage(2, p, stream);

  // MIX: X -> Y (b,co,dd32,hh32,ww16), per-mode 64x64 channel matrices
  mix_kernel<<<1024, 256, 0, stream>>>(Xb, w1, w2, w3, w4, Yb);

  // I1: Y -> G1 (b,co,dd,hh,w'33) complex (L=33). rows=(b,co,dd,hh)=131072
  p = {Yb, nullptr, G1re, G1im, tw + 3 * 8192,
       16, 48, 33, 8192, 3, 1, 16, 1, 1, 33, 1, 1.0f};
  launch_stage(0, p, stream);

  // I2: G1 -> G2 (b,co,dd,h64,w'33) complex. rows=(b,co,dd,w')=135168
  p = {G1re, G1im, G2re, G2im, tw + 4 * 8192,
       32, 64, 64, 8448, 4, 33, 1056, 33, 33, 2112, 33, 1.0f};
  launch_stage(1, p, stream);

  // I3: G2 -> out (b,co,d64,h,w') real (Re-Im)/135168. rows=(b,co,h,w')=270336
  p = {G2re, G2im, out, nullptr, tw + 5 * 8192,
       32, 64, 64, 16896, 4, 2112, 67584, 2112, 2112, 135168, 2112,
       1.0f / 135168.0f};
  launch_stage(2, p, stream);
}